// INECN_Layer_81088982548657
// MI455X (gfx1250) — compile-verified
//
#include <hip/hip_runtime.h>
#include <hip/hip_bf16.h>

typedef float v2f __attribute__((ext_vector_type(2)));
typedef float v4f __attribute__((ext_vector_type(4)));
typedef float v8f __attribute__((ext_vector_type(8)));

#define BM 128
#define BN 128
#define BK 32
// k-paired LDS layout: element (k, d) at (k>>1)*PAIR_STRIDE + d*2 + (k&1)
// -> fragment {lds[k][d], lds[k+1][d]} is one aligned 8-byte ds_load_b64.
#define PAIR_STRIDE (2 * (128 + 1))
#define LDS_FLOATS ((BK / 2) * PAIR_STRIDE)

// ---------------------------------------------------------------------------
// Generic fp32 WMMA GEMM:  C[M x N] = A_eff[M x K] * B_eff[K x N]
//  TA == true : A stored row-major [M][K]      (transpose-stage into LDS)
//  TA == false: A stored K-major   [K][M]      (direct-stage)
//  TB likewise for B.
//  fp32 16x16x4 fragment layout: lanes 0-15 -> K=0,1 ; lanes 16-31 -> K=2,3.
//  EPI==1: C = relu(acc + bias1[col][row] + bias2[row][col])
// ---------------------------------------------------------------------------
template <bool TA, bool TB, int EPI>
__global__ __launch_bounds__(256) void wmma_gemm_f32(
    const float* __restrict__ A, const float* __restrict__ B,
    float* __restrict__ C, int Mdim, int Ndim, int Kdim,
    long lda, long ldb, long ldc,
    const float* __restrict__ bias1, long ldb1,
    const float* __restrict__ bias2, long ldb2) {
  __shared__ float lsA[LDS_FLOATS];
  __shared__ float lsB[LDS_FLOATS];

  const int tid  = threadIdx.x;
  const int wid  = tid >> 5;          // 8 waves
  const int lane = tid & 31;          // wave32
  const int la   = lane & 15;
  const int khi  = (lane >> 4) << 1;  // 0 or 2
  const int wm   = (wid & 3) * 32;    // wave M offset within tile
  const int wn   = (wid >> 2) * 64;   // wave N offset within tile

  const int m0 = blockIdx.y * BM;
  const int n0 = blockIdx.x * BN;

  v8f acc[2][4];
#pragma unroll
  for (int i = 0; i < 2; ++i)
#pragma unroll
    for (int j = 0; j < 4; ++j) acc[i][j] = v8f{};

  for (int k0 = 0; k0 < Kdim; k0 += BK) {
    __syncthreads();
    // ---- stage A tile: (kk, mm) <- A_eff[m0+mm][k0+kk] ----
    if (TA) {  // memory [dim][K]: 16B loads along K, two b64 LDS stores
      for (int idx = tid; idx < (BK / 4) * 128; idx += 256) {
        int k4 = (idx & (BK / 4 - 1)) * 4;
        int dd = idx >> 3;
        v4f val = *(const v4f*)&A[(size_t)(m0 + dd) * lda + (k0 + k4)];
        float* p = &lsA[(k4 >> 1) * PAIR_STRIDE + dd * 2];
        *(v2f*)&p[0] = v2f{val.x, val.y};
        *(v2f*)&p[PAIR_STRIDE] = v2f{val.z, val.w};
      }
    } else {   // memory [K][dim]: 16B loads along dim, stride-2 LDS stores
      for (int idx = tid; idx < BK * (128 / 4); idx += 256) {
        int d4 = (idx & 31) * 4;
        int kk = idx >> 5;
        v4f val = *(const v4f*)&A[(size_t)(k0 + kk) * lda + (m0 + d4)];
        float* p = &lsA[(kk >> 1) * PAIR_STRIDE + (kk & 1)];
        p[(d4 + 0) * 2] = val.x;
        p[(d4 + 1) * 2] = val.y;
        p[(d4 + 2) * 2] = val.z;
        p[(d4 + 3) * 2] = val.w;
      }
    }
    // ---- stage B tile: (kk, nn) <- B_eff[k0+kk][n0+nn] ----
    if (TB) {
      for (int idx = tid; idx < (BK / 4) * 128; idx += 256) {
        int k4 = (idx & (BK / 4 - 1)) * 4;
        int dd = idx >> 3;
        v4f val = *(const v4f*)&B[(size_t)(n0 + dd) * ldb + (k0 + k4)];
        float* p = &lsB[(k4 >> 1) * PAIR_STRIDE + dd * 2];
        *(v2f*)&p[0] = v2f{val.x, val.y};
        *(v2f*)&p[PAIR_STRIDE] = v2f{val.z, val.w};
      }
    } else {
      for (int idx = tid; idx < BK * (128 / 4); idx += 256) {
        int d4 = (idx & 31) * 4;
        int kk = idx >> 5;
        v4f val = *(const v4f*)&B[(size_t)(k0 + kk) * ldb + (n0 + d4)];
        float* p = &lsB[(kk >> 1) * PAIR_STRIDE + (kk & 1)];
        p[(d4 + 0) * 2] = val.x;
        p[(d4 + 1) * 2] = val.y;
        p[(d4 + 2) * 2] = val.z;
        p[(d4 + 3) * 2] = val.w;
      }
    }
    __syncthreads();

#pragma unroll
    for (int kk = 0; kk < BK; kk += 4) {
      const int prow = (kk + khi) >> 1;  // exact: kk%4==0, khi in {0,2}
      v2f af[2], bf[4];
#pragma unroll
      for (int i = 0; i < 2; ++i)
        af[i] = *(const v2f*)&lsA[prow * PAIR_STRIDE + (wm + i * 16 + la) * 2];
#pragma unroll
      for (int j = 0; j < 4; ++j)
        bf[j] = *(const v2f*)&lsB[prow * PAIR_STRIDE + (wn + j * 16 + la) * 2];
#pragma unroll
      for (int i = 0; i < 2; ++i)
#pragma unroll
        for (int j = 0; j < 4; ++j)
          acc[i][j] = __builtin_amdgcn_wmma_f32_16x16x4_f32(
              false, af[i], false, bf[j], (short)0, acc[i][j], false, false);
    }
  }

  // ---- store: VGPR r -> M = r (lanes 0-15) / r+8 (lanes 16-31), N = la ----
  const int rbase = (lane >> 4) * 8;
#pragma unroll
  for (int i = 0; i < 2; ++i) {
#pragma unroll
    for (int j = 0; j < 4; ++j) {
#pragma unroll
      for (int r = 0; r < 8; ++r) {
        int row = m0 + wm + i * 16 + rbase + r;
        int col = n0 + wn + j * 16 + la;
        float v = acc[i][j][r];
        if (EPI == 1) {
          v += bias1[(size_t)col * ldb1 + row] + bias2[(size_t)row * ldb2 + col];
          v = fmaxf(v, 0.0f);
        }
        C[(size_t)row * ldc + col] = v;
      }
    }
  }
}

// ---------------------------------------------------------------------------
// Tw = W_ev @ W_ve (128x128), norm_XY = max abs row sum -> scal[0]
// ---------------------------------------------------------------------------
__global__ __launch_bounds__(128) void ww_norm_kernel(
    const float* __restrict__ Wev, const float* __restrict__ Wve,
    float* __restrict__ Tw, float* __restrict__ scal) {
  __shared__ float rowsum[128];
  int r = threadIdx.x;
  float asum = 0.0f;
  for (int c = 0; c < 128; ++c) {
    float s = 0.0f;
#pragma unroll 4
    for (int k = 0; k < 128; ++k) s += Wev[r * 128 + k] * Wve[k * 128 + c];
    Tw[r * 128 + c] = s;
    asum += fabsf(s);
  }
  rowsum[r] = asum;
  __syncthreads();
  if (r == 0) {
    float mx = rowsum[0];
    for (int i = 1; i < 128; ++i) mx = fmaxf(mx, rowsum[i]);
    scal[0] = mx;
  }
}

// ---------------------------------------------------------------------------
// Power iteration on PSD M (2048x2048) -> pf, then scale = min(1, sqrt(kappa/norm))
// Single persistent workgroup of 1024 threads; v lives in LDS.
// ---------------------------------------------------------------------------
__global__ __launch_bounds__(1024) void power_iter_kernel(
    const float* __restrict__ Mm, float* __restrict__ scal) {
  __shared__ float v[2048];
  __shared__ float red[1024];
  const int t = threadIdx.x;
  v[t] = 1.0f;
  v[t + 1024] = 1.0f;
  __syncthreads();
  float lam = 1.0f;
  for (int it = 0; it < 300; ++it) {
    const float* r0 = Mm + (size_t)(2 * t) * 2048;
    const float* r1 = r0 + 2048;
    float w0 = 0.0f, w1 = 0.0f;
#pragma unroll 4
    for (int j = 0; j < 2048; ++j) {
      float vj = v[j];
      w0 += r0[j] * vj;
      w1 += r1[j] * vj;
    }
    red[t] = w0 * w0 + w1 * w1;
    __syncthreads();  // all dot-products (reads of v) done
    for (int s = 512; s > 0; s >>= 1) {
      if (t < s) red[t] += red[t + s];
      __syncthreads();
    }
    float nrm = sqrtf(red[0]);
    float inv = 1.0f / nrm;
    v[2 * t] = w0 * inv;
    v[2 * t + 1] = w1 * inv;
    lam = nrm;  // ||M v_hat|| -> lambda_max as v converges
    __syncthreads();
  }
  if (t == 0) {
    float pf = lam;
    float norm_xy = scal[0];
    float kappa = 0.99f / pf;
    float sc = fminf(1.0f, sqrtf(kappa / norm_xy));
    scal[1] = pf;
    scal[2] = sc;
    scal[3] = sc * sc;
  }
}

// ---------------------------------------------------------------------------
// W_ev_s = s*W_ev ; W_ve_s = s*W_ve ; A_w = s^2 * Tw
// ---------------------------------------------------------------------------
__global__ __launch_bounds__(256) void scale_kernel(
    const float* __restrict__ Wev, const float* __restrict__ Wve,
    const float* __restrict__ Tw, float* __restrict__ Wev_s,
    float* __restrict__ Wve_s, float* __restrict__ Aw,
    const float* __restrict__ scal) {
  int i = blockIdx.x * 256 + threadIdx.x;
  float s = scal[2], s2 = scal[3];
  if (i < 128 * 128) {
    Wev_s[i] = Wev[i] * s;
    Wve_s[i] = Wve[i] * s;
    Aw[i] = Tw[i] * s2;
  }
}

// ---------------------------------------------------------------------------
extern "C" void kernel_launch(void* const* d_in, const int* in_sizes, int n_in,
                              void* d_out, int out_size, void* d_ws,
                              size_t ws_size, hipStream_t stream) {
  (void)in_sizes; (void)n_in; (void)out_size; (void)ws_size;
  const float* X0  = (const float*)d_in[0];  // (128, 2048)
  const float* H   = (const float*)d_in[1];  // (2048, 32768)
  const float* E0  = (const float*)d_in[2];  // (32768, 128)
  const float* F0  = (const float*)d_in[3];  // (128, 2048)
  const float* Wev = (const float*)d_in[4];  // (128, 128)
  const float* Wve = (const float*)d_in[5];  // (128, 128)

  float* out    = (float*)d_out;
  float* Ft_out = out;                 // 128*2048
  float* Et_out = out + 128 * 2048;    // 32768*128

  float* ws    = (float*)d_ws;
  float* Mm    = ws; ws += (size_t)2048 * 2048;  // H H^T
  float* Tw    = ws; ws += 128 * 128;            // W_ev W_ve (unscaled)
  float* Aw    = ws; ws += 128 * 128;            // scaled W_ev_s W_ve_s
  float* Wev_s = ws; ws += 128 * 128;
  float* Wve_s = ws; ws += 128 * 128;
  float* scal  = ws; ws += 16;                   // [normXY, pf, scale, scale2]
  float* G     = ws; ws += (size_t)32768 * 128;  // E0 @ W_ve_s
  float* BH    = ws; ws += (size_t)2048 * 128;   // H @ G   (bias, [n][p])
  float* Xa    = ws; ws += (size_t)128 * 2048;
  float* Xb    = ws; ws += (size_t)128 * 2048;
  float* Tbuf  = ws; ws += (size_t)128 * 2048;   // X @ M
  float* V     = ws; ws += (size_t)2048 * 128;   // Ft^T @ W_ev_s

  // 1) M = H @ H^T              (both operands [dim][K] -> transpose stage)
  wmma_gemm_f32<true, true, 0><<<dim3(2048 / BN, 2048 / BM), 256, 0, stream>>>(
      H, H, Mm, 2048, 2048, 32768, 32768, 32768, 2048, nullptr, 0, nullptr, 0);

  // 2) Tw = W_ev @ W_ve, norm_XY
  ww_norm_kernel<<<1, 128, 0, stream>>>(Wev, Wve, Tw, scal);

  // 3) power iteration -> pf, scale
  power_iter_kernel<<<1, 1024, 0, stream>>>(Mm, scal);

  // 4) scale W_ev, W_ve, A_w
  scale_kernel<<<64, 256, 0, stream>>>(Wev, Wve, Tw, Wev_s, Wve_s, Aw, scal);

  // 5) G = E0 @ W_ve_s          (A [dim][K] trans ; B [K][N] direct)
  wmma_gemm_f32<true, false, 0><<<dim3(1, 32768 / BM), 256, 0, stream>>>(
      E0, Wve_s, G, 32768, 128, 128, 128, 128, 128, nullptr, 0, nullptr, 0);

  // 6) BH = H @ G  (bias piece, shape [n=2048][p=128])
  wmma_gemm_f32<true, false, 0><<<dim3(1, 2048 / BM), 256, 0, stream>>>(
      H, G, BH, 2048, 128, 32768, 32768, 128, 128, nullptr, 0, nullptr, 0);

  // 7) fixed point: X <- relu(A_w @ (X @ M) + BH^T + F0), 50 iterations
  for (int i = 0; i < 50; ++i) {
    const float* src = (i == 0) ? X0 : ((i & 1) ? Xa : Xb);
    float* dst = (i == 49) ? Ft_out : ((i & 1) ? Xb : Xa);
    // T = X @ M   (A=X [128][2048] trans ; B=M [2048][2048] direct)
    wmma_gemm_f32<true, false, 0><<<dim3(2048 / BN, 1), 256, 0, stream>>>(
        src, Mm, Tbuf, 128, 2048, 2048, 2048, 2048, 2048, nullptr, 0, nullptr, 0);
    // X' = relu(A_w @ T + BH[col][row] + F0[row][col])
    wmma_gemm_f32<true, false, 1><<<dim3(2048 / BN, 1), 256, 0, stream>>>(
        Aw, Tbuf, dst, 128, 2048, 128, 128, 2048, 2048, BH, 128, F0, 2048);
  }

  // 8) V = Ft^T @ W_ev_s   (A_eff[n][p] = Ft[p][n] -> memory [K][dim] direct)
  wmma_gemm_f32<false, false, 0><<<dim3(1, 2048 / BM), 256, 0, stream>>>(
      Ft_out, Wev_s, V, 2048, 128, 128, 2048, 128, 128, nullptr, 0, nullptr, 0);

  // 9) Et = H^T @ V        (A_eff[m][n] = H[n][m] -> memory [K][dim] direct)
  wmma_gemm_f32<false, false, 0><<<dim3(1, 32768 / BM), 256, 0, stream>>>(
      H, V, Et_out, 32768, 128, 2048, 32768, 128, 128, nullptr, 0, nullptr, 0);
}